// GATModel_4587025072858
// MI455X (gfx1250) — compile-verified
//
#include <hip/hip_runtime.h>
#include <math.h>

// Problem constants (from reference)
#define Nn   50000
#define Ee   800000
#define ETot (Ee + Nn)   // edges + self loops = 850000
#define CINn 256
#define H1n  4
#define COUTn 64
#define HIDn 256          // H1*COUT
#define NCn  16

typedef __attribute__((ext_vector_type(2))) float v2f;
typedef __attribute__((ext_vector_type(8))) float v8f;

__device__ __forceinline__ float lrelu(float x) { return x > 0.f ? x : 0.2f * x; }

// float atomic max via signed/unsigned int punning (valid with -inf init)
__device__ __forceinline__ void atomicMaxF(float* addr, float v) {
    if (v >= 0.f) atomicMax((int*)addr, __float_as_int(v));
    else          atomicMin((unsigned int*)addr, __float_as_uint(v));
}

// ---------------------------------------------------------------- init
__global__ void k_init(float* out1, float* m1, float* d1,
                       float* out2, float* m2, float* d2) {
    int i = blockIdx.x * blockDim.x + threadIdx.x;
    int stride = gridDim.x * blockDim.x;
    for (int j = i; j < Nn * HIDn; j += stride) out1[j] = 0.f;
    for (int j = i; j < Nn * H1n;  j += stride) { m1[j] = -INFINITY; d1[j] = 0.f; }
    for (int j = i; j < Nn * NCn;  j += stride) out2[j] = 0.f;
    for (int j = i; j < Nn;        j += stride) { m2[j] = -INFINITY; d2[j] = 0.f; }
}

// ------------------------------------------------- layer-1 GEMM (WMMA f32)
// h1[n, f] = sum_k x[n,k] * W1flat[f*256 + k] + b1[f],  f = h*64 + o
// Block = 16 rows of x in LDS; 16 waves, each owns one 16-wide column tile.
__global__ void __launch_bounds__(512) k_gemm1(const float* __restrict__ x,
                                               const float* __restrict__ W1,
                                               const float* __restrict__ b1,
                                               float* __restrict__ h1) {
    __shared__ float xs[16 * 260];            // pad 256->260: conflict-free DS reads
    int row0 = blockIdx.x * 16;
    for (int idx = threadIdx.x; idx < 16 * 256; idx += 512) {
        int r = idx >> 8, c = idx & 255;
        int gr = row0 + r;
        xs[r * 260 + c] = (gr < Nn) ? x[(size_t)gr * CINn + c] : 0.f;
    }
    __syncthreads();

    int wave   = threadIdx.x >> 5;
    int lane   = threadIdx.x & 31;
    int laneLo = lane & 15;
    int hi     = lane >> 4;                   // 0: K pair {0,1}; 1: K pair {2,3}
    int f      = wave * 16 + laneLo;          // output feature column
    const float* wrow = W1 + (size_t)f * CINn;

    v8f acc = {};
#pragma unroll
    for (int t = 0; t < 64; ++t) {            // K = 256 = 64 * 4
        int kk = 4 * t + 2 * hi;
        v2f a; a[0] = xs[laneLo * 260 + kk]; a[1] = xs[laneLo * 260 + kk + 1];
        v2f b; b[0] = wrow[kk];              b[1] = wrow[kk + 1];
        acc = __builtin_amdgcn_wmma_f32_16x16x4_f32(
                  false, a, false, b, (short)0, acc, false, false);
    }
    float bias = b1[f];
#pragma unroll
    for (int j = 0; j < 8; ++j) {
        int r = row0 + j + 8 * hi;            // C/D layout: VGPR j -> rows j / j+8
        if (r < Nn) h1[(size_t)r * HIDn + f] = acc[j] + bias;
    }
}

// ------------------------------------------------- per-node head scores (L1)
__global__ void k_scores1(const float* __restrict__ h1,
                          const float* __restrict__ a11w, const float* __restrict__ a11b,
                          const float* __restrict__ a12w, const float* __restrict__ a12b,
                          float* __restrict__ s1, float* __restrict__ s2) {
    int tid = blockIdx.x * blockDim.x + threadIdx.x;
    if (tid >= Nn * H1n) return;
    int n = tid >> 2, h = tid & 3;
    const float* hp = h1 + (size_t)n * HIDn + h * COUTn;
    const float* w1 = a11w + h * COUTn;
    const float* w2 = a12w + h * COUTn;
    float acc1 = 0.f, acc2 = 0.f;
#pragma unroll 8
    for (int j = 0; j < COUTn; ++j) { float v = hp[j]; acc1 += v * w1[j]; acc2 += v * w2[j]; }
    s1[tid] = acc1 + a11b[h];
    s2[tid] = acc2 + a12b[h];
}

// ------------------------------------------------- layer-1 edge passes
__device__ __forceinline__ void edge_sd(const long long* ei, int e, long long& s, long long& d) {
    if (e < Ee) { s = ei[e]; d = ei[Ee + e]; } else { s = d = (long long)(e - Ee); }
}

__global__ void k_emax1(const long long* __restrict__ ei, const float* __restrict__ s1,
                        const float* __restrict__ s2, float* __restrict__ m1) {
    int e = blockIdx.x * blockDim.x + threadIdx.x;
    if (e >= ETot) return;
    long long s, d; edge_sd(ei, e, s, d);
#pragma unroll
    for (int h = 0; h < H1n; ++h)
        atomicMaxF(&m1[d * H1n + h], lrelu(s2[d * H1n + h] + s1[s * H1n + h]));
}

__global__ void k_eden1(const long long* __restrict__ ei, const float* __restrict__ s1,
                        const float* __restrict__ s2, const float* __restrict__ m1,
                        float* __restrict__ d1) {
    int e = blockIdx.x * blockDim.x + threadIdx.x;
    if (e >= ETot) return;
    long long s, d; edge_sd(ei, e, s, d);
#pragma unroll
    for (int h = 0; h < H1n; ++h) {
        float ev = lrelu(s2[d * H1n + h] + s1[s * H1n + h]);
        atomicAdd(&d1[d * H1n + h], __expf(ev - m1[d * H1n + h]));
    }
}

// one wave per edge: lanes cover 256 features coalesced
__global__ void __launch_bounds__(256) k_eaggr1(const long long* __restrict__ ei,
                        const float* __restrict__ s1, const float* __restrict__ s2,
                        const float* __restrict__ m1, const float* __restrict__ d1,
                        const float* __restrict__ h1, float* __restrict__ out1) {
    int e = blockIdx.x * 8 + (threadIdx.x >> 5);
    if (e >= ETot) return;
    int lane = threadIdx.x & 31;
    long long s, d; edge_sd(ei, e, s, d);
    float att[H1n];
#pragma unroll
    for (int h = 0; h < H1n; ++h) {
        float ev = lrelu(s2[d * H1n + h] + s1[s * H1n + h]);
        att[h] = __expf(ev - m1[d * H1n + h]) / (d1[d * H1n + h] + 1e-16f);
    }
    const float* hs = h1 + (size_t)s * HIDn;
    float* od = out1 + (size_t)d * HIDn;
#pragma unroll
    for (int j = 0; j < 8; ++j) {
        int f = lane + 32 * j;
        atomicAdd(&od[f], hs[f] * att[f >> 6]);
    }
}

// ------------------------------------------------- bias + ELU (writes into h1 buffer)
__global__ void k_elu(const float* __restrict__ out1, const float* __restrict__ bias1,
                      float* __restrict__ h2) {
    int i = blockIdx.x * blockDim.x + threadIdx.x;
    if (i >= Nn * HIDn) return;
    float v = out1[i] + bias1[i & 255];
    h2[i] = v > 0.f ? v : expm1f(v);
}

// ------------------------------------------------- layer-2 GEMM (WMMA f32), N=16 cols
__global__ void __launch_bounds__(256) k_gemm2(const float* __restrict__ h2,
                                               const float* __restrict__ W2,
                                               const float* __restrict__ b2,
                                               float* __restrict__ g) {
    int tile = blockIdx.x * 8 + (threadIdx.x >> 5);
    if (tile * 16 >= Nn) return;              // wave-uniform exit (EXEC stays full)
    int row0 = tile * 16;
    int lane = threadIdx.x & 31;
    int laneLo = lane & 15, hi = lane >> 4;
    int ra = row0 + laneLo;
    bool aval = ra < Nn;
    const float* arow = h2 + (size_t)(aval ? ra : (Nn - 1)) * HIDn;
    const float* wrow = W2 + (size_t)laneLo * CINn;   // f = laneLo (16 classes)

    v8f acc = {};
#pragma unroll
    for (int t = 0; t < 64; ++t) {
        int kk = 4 * t + 2 * hi;
        v2f a; a[0] = aval ? arow[kk] : 0.f; a[1] = aval ? arow[kk + 1] : 0.f;
        v2f b; b[0] = wrow[kk];              b[1] = wrow[kk + 1];
        acc = __builtin_amdgcn_wmma_f32_16x16x4_f32(
                  false, a, false, b, (short)0, acc, false, false);
    }
    float bias = b2[laneLo];
#pragma unroll
    for (int j = 0; j < 8; ++j) {
        int r = row0 + j + 8 * hi;
        if (r < Nn) g[(size_t)r * NCn + laneLo] = acc[j] + bias;
    }
}

// ------------------------------------------------- layer-2 scores + edge passes
__global__ void k_scores2(const float* __restrict__ g,
                          const float* __restrict__ a21w, const float* __restrict__ a21b,
                          const float* __restrict__ a22w, const float* __restrict__ a22b,
                          float* __restrict__ t1, float* __restrict__ t2) {
    int n = blockIdx.x * blockDim.x + threadIdx.x;
    if (n >= Nn) return;
    const float* gp = g + (size_t)n * NCn;
    float acc1 = 0.f, acc2 = 0.f;
#pragma unroll
    for (int j = 0; j < NCn; ++j) { float v = gp[j]; acc1 += v * a21w[j]; acc2 += v * a22w[j]; }
    t1[n] = acc1 + a21b[0];
    t2[n] = acc2 + a22b[0];
}

__global__ void k_emax2(const long long* __restrict__ ei, const float* __restrict__ t1,
                        const float* __restrict__ t2, float* __restrict__ m2) {
    int e = blockIdx.x * blockDim.x + threadIdx.x;
    if (e >= ETot) return;
    long long s, d; edge_sd(ei, e, s, d);
    atomicMaxF(&m2[d], lrelu(t2[d] + t1[s]));
}

__global__ void k_eden2(const long long* __restrict__ ei, const float* __restrict__ t1,
                        const float* __restrict__ t2, const float* __restrict__ m2,
                        float* __restrict__ d2) {
    int e = blockIdx.x * blockDim.x + threadIdx.x;
    if (e >= ETot) return;
    long long s, d; edge_sd(ei, e, s, d);
    float ev = lrelu(t2[d] + t1[s]);
    atomicAdd(&d2[d], __expf(ev - m2[d]));
}

__global__ void k_eaggr2(const long long* __restrict__ ei, const float* __restrict__ t1,
                         const float* __restrict__ t2, const float* __restrict__ m2,
                         const float* __restrict__ d2, const float* __restrict__ g,
                         float* __restrict__ out2) {
    int e = blockIdx.x * blockDim.x + threadIdx.x;
    if (e >= ETot) return;
    long long s, d; edge_sd(ei, e, s, d);
    float ev = lrelu(t2[d] + t1[s]);
    float att = __expf(ev - m2[d]) / (d2[d] + 1e-16f);
    const float* gs = g + (size_t)s * NCn;
    float* od = out2 + (size_t)d * NCn;
#pragma unroll
    for (int j = 0; j < NCn; ++j) atomicAdd(&od[j], gs[j] * att);
}

// ------------------------------------------------- bias + log_softmax
__global__ void k_final(const float* __restrict__ out2, const float* __restrict__ bias2,
                        float* __restrict__ y) {
    int n = blockIdx.x * blockDim.x + threadIdx.x;
    if (n >= Nn) return;
    float v[NCn]; float mx = -INFINITY;
#pragma unroll
    for (int j = 0; j < NCn; ++j) { v[j] = out2[(size_t)n * NCn + j] + bias2[j]; mx = fmaxf(mx, v[j]); }
    float sum = 0.f;
#pragma unroll
    for (int j = 0; j < NCn; ++j) sum += __expf(v[j] - mx);
    float l = logf(sum);
#pragma unroll
    for (int j = 0; j < NCn; ++j) y[(size_t)n * NCn + j] = v[j] - mx - l;
}

// ------------------------------------------------- host launch
extern "C" void kernel_launch(void* const* d_in, const int* in_sizes, int n_in,
                              void* d_out, int out_size, void* d_ws, size_t ws_size,
                              hipStream_t stream) {
    const float*     x     = (const float*)d_in[0];
    const long long* ei    = (const long long*)d_in[1];   // jnp.int64 edge_index (2,E)
    const float*     W1    = (const float*)d_in[2];
    const float*     b1    = (const float*)d_in[3];
    const float*     a11w  = (const float*)d_in[4];
    const float*     a11b  = (const float*)d_in[5];
    const float*     a12w  = (const float*)d_in[6];
    const float*     a12b  = (const float*)d_in[7];
    const float*     bias1 = (const float*)d_in[8];
    const float*     W2    = (const float*)d_in[9];
    const float*     b2    = (const float*)d_in[10];
    const float*     a21w  = (const float*)d_in[11];
    const float*     a21b  = (const float*)d_in[12];
    const float*     a22w  = (const float*)d_in[13];
    const float*     a22b  = (const float*)d_in[14];
    const float*     bias2 = (const float*)d_in[15];

    float* ws   = (float*)d_ws;
    float* h1   = ws;                              // N*256 (reused for post-ELU h)
    float* out1 = h1   + (size_t)Nn * HIDn;        // N*256
    float* s1   = out1 + (size_t)Nn * HIDn;        // N*4
    float* s2   = s1   + (size_t)Nn * H1n;
    float* m1   = s2   + (size_t)Nn * H1n;
    float* d1   = m1   + (size_t)Nn * H1n;
    float* g    = d1   + (size_t)Nn * H1n;         // N*16
    float* out2 = g    + (size_t)Nn * NCn;         // N*16
    float* t1   = out2 + (size_t)Nn * NCn;         // N
    float* t2   = t1   + Nn;
    float* m2   = t2   + Nn;
    float* d2   = m2   + Nn;

    k_init   <<<1024, 256, 0, stream>>>(out1, m1, d1, out2, m2, d2);
    k_gemm1  <<<(Nn + 15) / 16, 512, 0, stream>>>(x, W1, b1, h1);
    k_scores1<<<(Nn * H1n + 255) / 256, 256, 0, stream>>>(h1, a11w, a11b, a12w, a12b, s1, s2);
    k_emax1  <<<(ETot + 255) / 256, 256, 0, stream>>>(ei, s1, s2, m1);
    k_eden1  <<<(ETot + 255) / 256, 256, 0, stream>>>(ei, s1, s2, m1, d1);
    k_eaggr1 <<<(ETot + 7) / 8, 256, 0, stream>>>(ei, s1, s2, m1, d1, h1, out1);
    k_elu    <<<(Nn * HIDn + 255) / 256, 256, 0, stream>>>(out1, bias1, h1);
    k_gemm2  <<<(((Nn + 15) / 16) + 7) / 8, 256, 0, stream>>>(h1, W2, b2, g);
    k_scores2<<<(Nn + 255) / 256, 256, 0, stream>>>(g, a21w, a21b, a22w, a22b, t1, t2);
    k_emax2  <<<(ETot + 255) / 256, 256, 0, stream>>>(ei, t1, t2, m2);
    k_eden2  <<<(ETot + 255) / 256, 256, 0, stream>>>(ei, t1, t2, m2, d2);
    k_eaggr2 <<<(ETot + 255) / 256, 256, 0, stream>>>(ei, t1, t2, m2, d2, g, out2);
    k_final  <<<(Nn + 255) / 256, 256, 0, stream>>>(out2, bias2, (float*)d_out);
}